// FNO1dBlock_39951785787474
// MI455X (gfx1250) — compile-verified
//
#include <hip/hip_runtime.h>
#include <cstdint>

// ---------------------------------------------------------------------------
// FNO1d block for MI455X (gfx1250): only 64 spectral modes are live, so the
// rfft/irfft collapse to two trig-basis GEMMs done with V_WMMA_F32_16X16X4_F32.
// L = 2^18  ->  frac(k*n/L) = ((k*n) & (L-1)) * (1/L) exactly (int arithmetic),
// and v_cos_f32 / v_sin_f32 take input in revolutions: cos(2*pi*t).
//
// Roofline: 128 MB traffic (~5.5us @ 23.3 TB/s), 8.6 GFLOP fp32 via WMMA
// (~4.2M v_wmma_f32_16x16x4_f32, 2048 FLOP each). Trig basis is generated by
// a complex-rotation recurrence (4 mul + 4 fma per k-step) so inner loops are
// WMMA + FMA + ds_load only; transcendentals appear only at init.
// Forward kernel double-buffers its 16 KB slabs through the async-LDS engine
// (ASYNCcnt) so the DMA overlaps the WMMA stream.
// ---------------------------------------------------------------------------

#define L_SIZE    262144
#define LMASK     (L_SIZE - 1)
#define CCH       64
#define NMODES    64

#define FWD_BLOCKS 256
#define FWD_CHUNK  (L_SIZE / FWD_BLOCKS)   // 1024 rows per block
#define STAGE_ROWS 64                      // 64 rows * 64 ch * 4B = 16 KB slab
#define NSTAGE     (FWD_CHUNK / STAGE_ROWS)

__constant__ const float kInvL = 1.0f / (float)L_SIZE;

typedef __attribute__((ext_vector_type(2))) float v2f;
typedef __attribute__((ext_vector_type(8))) float v8f;
typedef __attribute__((ext_vector_type(4))) int   v4i;

// hardware trig: input in revolutions, computes cos/sin(2*pi*t)
__device__ __forceinline__ float cos2pi(float t) { return __builtin_amdgcn_cosf(t); }
__device__ __forceinline__ float sin2pi(float t) { return __builtin_amdgcn_sinf(t); }

// D = A(16x4 f32) * B(4x16 f32) + C(16x16 f32)
__device__ __forceinline__ v8f wmma_f32(v2f a, v2f b, v8f c) {
  return __builtin_amdgcn_wmma_f32_16x16x4_f32(false, a, false, b, (short)0, c,
                                               false, false);
}

#if defined(__has_builtin)
#if __has_builtin(__builtin_amdgcn_global_load_async_to_lds_b128)
#define USE_ASYNC_LDS 1
#endif
#endif

// async-LDS builtin takes (v4i __device__*, v4i __shared__*, imm off, imm cpol)
typedef __attribute__((address_space(1))) v4i as1_v4i;
typedef __attribute__((address_space(3))) v4i as3_v4i;

// ---------------------------------------------------------------------------
// Kernel 0: zero the Ur/Us accumulators in workspace (ws is poisoned by harness)
// ---------------------------------------------------------------------------
__global__ __launch_bounds__(256) void fno_zero(float* __restrict__ p, int n) {
  int i = blockIdx.x * 256 + threadIdx.x;
  if (i < n) p[i] = 0.0f;
}

// ---------------------------------------------------------------------------
// Kernel 1: forward modes.
//   Ur[k,i] = sum_n u[n,i] * cos(2*pi*k*n/L)
//   Us[k,i] = sum_n u[n,i] * sin(2*pi*k*n/L)     (u_hat = Ur - i*Us)
// 256 blocks, each reduces a 1024-row chunk via WMMA; double-buffered async
// LDS staging; atomic-add of 64x64 partials at the end.
// ---------------------------------------------------------------------------
__global__ __launch_bounds__(256) void fno_forward(const float* __restrict__ u,
                                                   float* __restrict__ Ur,
                                                   float* __restrict__ Us) {
  __shared__ float lds[2][STAGE_ROWS * CCH];  // 2 x 16 KB slabs

  const int tid    = threadIdx.x;
  const int wave   = tid >> 5;
  const int lane   = tid & 31;
  const int laneLo = lane & 15;
  const int hiSel  = (lane >> 4) & 1;      // 0: lanes 0-15, 1: lanes 16-31

  const int kTile   = wave >> 1;           // mode tile 0..3
  const int iTile0  = (wave & 1) * 2;      // channel tile pair {0,1} or {2,3}
  const int modeRow = kTile * 16 + laneLo; // A-matrix M index (mode)

  const int chunkBase = blockIdx.x * FWD_CHUNK;

  // --- trig recurrence init: angle theta(n) = 2*pi*(modeRow*n mod L)/L,
  //     n advances by 4 per k-step across the whole chunk. Rotation constant:
  const float td = (float)(((unsigned)modeRow * 4u) & LMASK) * kInvL;
  const float cd = cos2pi(td), sd = sin2pi(td);
  float c0, s0, c1, s1;  // (cos,sin) at n0 and n0+1
  {
    const unsigned n0 = (unsigned)(chunkBase + hiSel * 2);
    const float t0 = (float)(((unsigned)modeRow * n0) & LMASK) * kInvL;
    const float t1 = (float)(((unsigned)modeRow * (n0 + 1u)) & LMASK) * kInvL;
    c0 = cos2pi(t0); s0 = sin2pi(t0);
    c1 = cos2pi(t1); s1 = sin2pi(t1);
  }

  v8f accC0 = {}; v8f accC1 = {};          // cos accumulators (2 channel tiles)
  v8f accS0 = {}; v8f accS1 = {};          // sin accumulators

  auto issue_stage = [&](int st, int bufIdx) {
    const float* gsrc = u + (size_t)(chunkBase + st * STAGE_ROWS) * CCH;
    float* dst = lds[bufIdx];
#ifdef USE_ASYNC_LDS
    for (int v = tid; v < STAGE_ROWS * CCH / 4; v += 256)
      __builtin_amdgcn_global_load_async_to_lds_b128(
          (as1_v4i*)((const float4*)gsrc + v),
          (as3_v4i*)((float4*)dst + v), 0, 0);
#else
    __builtin_prefetch(gsrc, 0, 1);
    for (int v = tid; v < STAGE_ROWS * CCH / 4; v += 256)
      ((float4*)dst)[v] = ((const float4*)gsrc)[v];
#endif
  };

  issue_stage(0, 0);

  for (int s = 0; s < NSTAGE; ++s) {
    const int cur = s & 1;
#ifdef USE_ASYNC_LDS
    asm volatile("s_wait_asynccnt 0x0" ::: "memory");  // own async loads done
#endif
    __syncthreads();  // everyone's slab written; prev compute on 1-cur finished
    if (s + 1 < NSTAGE) issue_stage(s + 1, 1 - cur);   // overlap DMA w/ compute

    const float* buf = lds[cur];
    for (int kk = 0; kk < STAGE_ROWS; kk += 4) {
      // ---- A fragments: lane layout M=laneLo; v0<->K=hiSel*2, v1<->K+1 ----
      v2f aC, aS;
      aC.x = c0; aC.y = c1;
      aS.x = s0; aS.y = s1;
      // rotate by delta (n += 4): cos(t+d)=c*cd-s*sd, sin(t+d)=s*cd+c*sd
      {
        const float nc0 = c0 * cd - s0 * sd, ns0 = s0 * cd + c0 * sd;
        const float nc1 = c1 * cd - s1 * sd, ns1 = s1 * cd + c1 * sd;
        c0 = nc0; s0 = ns0; c1 = nc1; s1 = ns1;
      }

      // ---- B fragments from LDS: 4 rows x 16 channels ----
      const int r0 = kk + hiSel * 2;
      v2f b0, b1;
      b0.x = buf[r0 * CCH       + iTile0 * 16 + laneLo];
      b0.y = buf[(r0 + 1) * CCH + iTile0 * 16 + laneLo];
      b1.x = buf[r0 * CCH       + (iTile0 + 1) * 16 + laneLo];
      b1.y = buf[(r0 + 1) * CCH + (iTile0 + 1) * 16 + laneLo];

      accC0 = wmma_f32(aC, b0, accC0);
      accC1 = wmma_f32(aC, b1, accC1);
      accS0 = wmma_f32(aS, b0, accS0);
      accS1 = wmma_f32(aS, b1, accS1);
    }
  }

  // C/D layout: VGPR r -> M = r (lanes 0-15) or r+8 (lanes 16-31); N = laneLo
  for (int r = 0; r < 8; ++r) {
    const int k  = kTile * 16 + (hiSel ? r + 8 : r);
    const int i0 = iTile0 * 16 + laneLo;
    atomicAdd(&Ur[k * CCH + i0],      accC0[r]);
    atomicAdd(&Ur[k * CCH + i0 + 16], accC1[r]);
    atomicAdd(&Us[k * CCH + i0],      accS0[r]);
    atomicAdd(&Us[k * CCH + i0 + 16], accS1[r]);
  }
}

// ---------------------------------------------------------------------------
// Kernel 2: combine with Kc, fold irfft weights + z_local.
//   u_hat = Ur - i*Us,  Kc = K0 + i*K1
//   zr[x,o] = sum_i Ur*K0 + Us*K1 ;  zi[x,o] = sum_i Ur*K1 - Us*K0
//   s[n,o] = (1/L)[ zr[0,o] + 2*sum_{k>=1}( zr cos - zi sin ) ]  (irfft drops zi[0])
//   Ar[k,o] = (k==0?1:2)/L * zr ;  As[k,o] = (k==0?0:-2/L) * zi
//   Ar[0,o] += z_local[o]   (cos(0)=1 broadcasts z_local over all rows)
// ---------------------------------------------------------------------------
__global__ __launch_bounds__(256) void fno_combine(
    const float* __restrict__ Ur, const float* __restrict__ Us,
    const float* __restrict__ K, const float* __restrict__ W,
    const float* __restrict__ u, const float* __restrict__ y,
    float* __restrict__ Ar, float* __restrict__ As) {
  const int tid = threadIdx.x;
  const float* K0 = K;
  const float* K1 = K + 64 * 64 * 64;

  for (int p = tid; p < NMODES * CCH; p += 256) {
    const int x = p >> 6;   // mode
    const int o = p & 63;   // out channel
    float zr = 0.0f, zi = 0.0f;
    for (int i = 0; i < CCH; ++i) {
      const float ur = Ur[x * CCH + i];
      const float us = Us[x * CCH + i];
      const float k0 = K0[(i * 64 + o) * 64 + x];
      const float k1 = K1[(i * 64 + o) * 64 + x];
      zr += ur * k0 + us * k1;
      zi += ur * k1 - us * k0;
    }
    const float wr = (x == 0) ? kInvL : 2.0f * kInvL;
    Ar[p] = wr * zr;
    As[p] = (x == 0) ? 0.0f : (-2.0f * kInvL) * zi;
  }
  __syncthreads();

  if (tid < CCH) {
    int idx = (int)(y[0] * (float)L_SIZE);   // jnp.int32 truncation
    idx = idx < 0 ? 0 : (idx > L_SIZE - 1 ? L_SIZE - 1 : idx);
    float zl = 0.0f;
    for (int i = 0; i < CCH; ++i) zl += W[tid * CCH + i] * u[(size_t)idx * CCH + i];
    Ar[tid] += zl;  // row k==0
  }
}

// ---------------------------------------------------------------------------
// Kernel 3: inverse transform. s[n,o] = sum_k Ar[k,o]*cos + As[k,o]*sin.
// (L x 128) x (128 x 64) GEMM; coef = [Ar ; As] staged once in 32 KB LDS.
// 4096 blocks x 64 rows; 8 waves x 2 tiles x 32 K-steps = 512 WMMA / block.
// The mode sequence restarts at the sin half (coef rows 64..127), so the same
// rotation recurrence is replayed from saved initial values, taking sin.
// ---------------------------------------------------------------------------
__global__ __launch_bounds__(256) void fno_inverse(const float* __restrict__ coefG,
                                                   float* __restrict__ out) {
  __shared__ float coef[128 * CCH];  // rows 0-63 = Ar, rows 64-127 = As (32 KB)

  const int tid = threadIdx.x;
  for (int v = tid; v < 128 * CCH / 4; v += 256)
    ((float4*)coef)[v] = ((const float4*)coefG)[v];
  __syncthreads();

  const int wave   = tid >> 5;
  const int lane   = tid & 31;
  const int laneLo = lane & 15;
  const int hiSel  = (lane >> 4) & 1;

  const int rowTile = wave & 3;          // 0..3 -> rows
  const int colT0   = (wave >> 2) * 2;   // 0 or 2 -> channel tile pair
  const int n       = blockIdx.x * 64 + rowTile * 16 + laneLo;  // output row

  // trig recurrence: theta(mode) = 2*pi*(mode*n mod L)/L, mode advances by 4.
  const float td = (float)(((unsigned)n * 4u) & LMASK) * kInvL;
  const float cd = cos2pi(td), sd = sin2pi(td);
  const unsigned mode0 = (unsigned)(hiSel * 2);
  float ic0, is0, ic1, is1;  // initial (cos,sin) at mode0 and mode0+1
  {
    const float t0 = (float)((mode0 * (unsigned)n) & LMASK) * kInvL;
    const float t1 = (float)(((mode0 + 1u) * (unsigned)n) & LMASK) * kInvL;
    ic0 = cos2pi(t0); is0 = sin2pi(t0);
    ic1 = cos2pi(t1); is1 = sin2pi(t1);
  }

  v8f acc0 = {}; v8f acc1 = {};

  for (int phase = 0; phase < 2; ++phase) {   // 0: cos half, 1: sin half
    float c0 = ic0, s0 = is0, c1 = ic1, s1 = is1;
    const int rowBase = phase * 64;
    for (int ks = 0; ks < 64; ks += 4) {
      const int b0 = rowBase + ks + hiSel * 2;  // coef row of v0

      v2f a;
      if (phase == 0) { a.x = c0; a.y = c1; }
      else            { a.x = s0; a.y = s1; }
      {
        const float nc0 = c0 * cd - s0 * sd, ns0 = s0 * cd + c0 * sd;
        const float nc1 = c1 * cd - s1 * sd, ns1 = s1 * cd + c1 * sd;
        c0 = nc0; s0 = ns0; c1 = nc1; s1 = ns1;
      }

      v2f bA, bB;
      bA.x = coef[b0 * CCH       + colT0 * 16 + laneLo];
      bA.y = coef[(b0 + 1) * CCH + colT0 * 16 + laneLo];
      bB.x = coef[b0 * CCH       + (colT0 + 1) * 16 + laneLo];
      bB.y = coef[(b0 + 1) * CCH + (colT0 + 1) * 16 + laneLo];

      acc0 = wmma_f32(a, bA, acc0);
      acc1 = wmma_f32(a, bB, acc1);
    }
  }

  for (int r = 0; r < 8; ++r) {
    const int row = blockIdx.x * 64 + rowTile * 16 + (hiSel ? r + 8 : r);
    const int c0_ = colT0 * 16 + laneLo;
    out[(size_t)row * CCH + c0_]      = acc0[r];
    out[(size_t)row * CCH + c0_ + 16] = acc1[r];
  }
}

// ---------------------------------------------------------------------------
// Launcher. Inputs: u (L*C f32), y (1 f32), K (2*64*64*64 f32), W (64*64 f32).
// Workspace layout (floats): [0,4096) Ur | [4096,8192) Us |
//                            [8192,12288) Ar | [12288,16384) As  (Ar,As contiguous)
// ---------------------------------------------------------------------------
extern "C" void kernel_launch(void* const* d_in, const int* in_sizes, int n_in,
                              void* d_out, int out_size, void* d_ws, size_t ws_size,
                              hipStream_t stream) {
  const float* u = (const float*)d_in[0];
  const float* y = (const float*)d_in[1];
  const float* K = (const float*)d_in[2];
  const float* W = (const float*)d_in[3];
  float* out = (float*)d_out;
  float* ws  = (float*)d_ws;

  float* Ur = ws;
  float* Us = ws + 4096;
  float* Ar = ws + 8192;
  float* As = ws + 12288;

  fno_zero<<<(8192 + 255) / 256, 256, 0, stream>>>(Ur, 8192);
  fno_forward<<<FWD_BLOCKS, 256, 0, stream>>>(u, Ur, Us);
  fno_combine<<<1, 256, 0, stream>>>(Ur, Us, K, W, u, y, Ar, As);
  fno_inverse<<<L_SIZE / 64, 256, 0, stream>>>(Ar, out);
}